// CosineSimilarity_76785425318088
// MI455X (gfx1250) — compile-verified
//
#include <hip/hip_runtime.h>
#include <stdint.h>

// ---------------------------------------------------------------------------
// Cosine similarity per row: out[i] = (p_i . h_i) / (max(||p_i||,eps)*max(||h_i||,eps))
// N=65536 rows, D=1024 fp32. Memory-bound (0.75 FLOP/byte): optimized as a
// TDM-fed (tensor_load_to_lds) double-buffered streaming reduction.
// ---------------------------------------------------------------------------

typedef __attribute__((ext_vector_type(4))) unsigned int tdm_v4u;
typedef __attribute__((ext_vector_type(8))) int          tdm_v8i;
typedef __attribute__((ext_vector_type(4))) int          tdm_v4i;

#define DCOLS     1024          // row length (fp32)
#define TILE_ROWS 4             // rows per TDM tile (16 KiB per tensor per tile)
#define BLOCK     128           // 4 wave32 waves: one wave per tile row

// Issue one TDM descriptor: load TILE_ROWS x DCOLS fp32 (row-major, stride
// DCOLS) from gsrc into LDS at byte offset lds_byte_off. Tracked on TENSORcnt.
__device__ __forceinline__ void tdm_load_tile(const float* __restrict__ gsrc,
                                              uint32_t lds_byte_off) {
  const uint64_t ga = (uint64_t)(uintptr_t)gsrc;
  tdm_v4u g0;
  g0[0] = 1u;                                    // [1:0] count=1 (valid user D#)
  g0[1] = lds_byte_off;                          // [63:32] lds_addr (bytes)
  g0[2] = (uint32_t)ga;                          // [95:64] global_addr lo
  g0[3] = ((uint32_t)(ga >> 32) & 0x01FFFFFFu)   // [120:96] global_addr[56:32]
        | 0x80000000u;                           // [127:126] type=2 ("image")
  tdm_v8i g1;
  g1[0] = (int)(2u << 16);                       // data_size=2 -> 4-byte elems; wg_mask=0
  g1[1] = (int)((DCOLS & 0xFFFFu) << 16);        // tensor_dim0[15:0] in [31:16]
  g1[2] = (int)((TILE_ROWS & 0xFFFFu) << 16);    // tensor_dim0 hi=0 | tensor_dim1[15:0]
  g1[3] = (int)((DCOLS & 0xFFFFu) << 16);        // tensor_dim1 hi=0 | tile_dim0
  g1[4] = (int)(TILE_ROWS & 0xFFFFu);            // tile_dim1 | tile_dim2=0
  g1[5] = (int)DCOLS;                            // tensor_dim0_stride[31:0]
  g1[6] = 0;                                     // stride hi | dim1_stride lo (unused, 2D)
  g1[7] = 0;
  tdm_v4i z4 = (tdm_v4i)0;                       // group 2/3 unused (<=2D tensor)
  tdm_v8i z8 = (tdm_v8i)0;                       // extra group (6-arg toolchain form)
  __builtin_amdgcn_tensor_load_to_lds(g0, g1, z4, z4, z8, 0);
}

__global__ void __launch_bounds__(BLOCK)
CosineSimilarity_76785425318088_kernel(const float* __restrict__ p,
                                       const float* __restrict__ h,
                                       float* __restrict__ out,
                                       int tiles) {
  // [buf 0/1][tensor p/h][row][col] : 2*2*4*1024 fp32 = 64 KiB LDS
  __shared__ __align__(16) float smem[2 * 2 * TILE_ROWS * DCOLS];

  const int  wave   = (int)(threadIdx.x >> 5);
  const int  lane   = (int)(threadIdx.x & 31);
  const bool issuer = (threadIdx.x < 32);   // wave 0 drives the TDM (EXEC ignored)

  int t = (int)blockIdx.x;
  if (t >= tiles) return;                   // uniform per block

  auto buf_off = [&](int buf, int which) -> uint32_t {
    const float* ptr = &smem[((buf * 2 + which) * TILE_ROWS) * DCOLS];
    // generic pointer low 32 bits == wave-relative LDS byte offset
    return (uint32_t)(uintptr_t)ptr;
  };

  // Prologue: start DMA of the first tile into buffer 0.
  if (issuer) {
    tdm_load_tile(p + (size_t)t * TILE_ROWS * DCOLS, buf_off(0, 0));
    tdm_load_tile(h + (size_t)t * TILE_ROWS * DCOLS, buf_off(0, 1));
  }

  int buf = 0;
  for (; t < tiles; t += (int)gridDim.x) {
    const int  nt        = t + (int)gridDim.x;
    const bool have_next = (nt < tiles);

    if (issuer) {
      if (have_next) {
        // Kick off next tile into the other buffer, then wait for current.
        tdm_load_tile(p + (size_t)nt * TILE_ROWS * DCOLS, buf_off(buf ^ 1, 0));
        tdm_load_tile(h + (size_t)nt * TILE_ROWS * DCOLS, buf_off(buf ^ 1, 1));
        // TDM ops from one wave complete in order: <=2 outstanding means the
        // two descriptors of the *current* tile have landed in LDS.
        __builtin_amdgcn_s_wait_tensorcnt(2);
      } else {
        __builtin_amdgcn_s_wait_tensorcnt(0);
      }
    }
    __syncthreads();   // publish LDS tile to all 4 waves

    // Each wave reduces one row: 8 x (2x ds_load_b128 + 12 FMA) per lane.
    const float4* pr = reinterpret_cast<const float4*>(
        &smem[((buf * 2 + 0) * TILE_ROWS + wave) * DCOLS]);
    const float4* hr = reinterpret_cast<const float4*>(
        &smem[((buf * 2 + 1) * TILE_ROWS + wave) * DCOLS]);

    float pp = 0.f, hh = 0.f, ph = 0.f;
#pragma unroll
    for (int k = 0; k < DCOLS / 128; ++k) {
      const float4 a = pr[k * 32 + lane];
      const float4 b = hr[k * 32 + lane];
      pp = fmaf(a.x, a.x, fmaf(a.y, a.y, fmaf(a.z, a.z, fmaf(a.w, a.w, pp))));
      hh = fmaf(b.x, b.x, fmaf(b.y, b.y, fmaf(b.z, b.z, fmaf(b.w, b.w, hh))));
      ph = fmaf(a.x, b.x, fmaf(a.y, b.y, fmaf(a.z, b.z, fmaf(a.w, b.w, ph))));
    }

    // wave32 butterfly reduction
#pragma unroll
    for (int m = 16; m > 0; m >>= 1) {
      pp += __shfl_xor(pp, m, 32);
      hh += __shfl_xor(hh, m, 32);
      ph += __shfl_xor(ph, m, 32);
    }

    if (lane == 0) {
      const float denom = fmaxf(sqrtf(pp), 1e-12f) * fmaxf(sqrtf(hh), 1e-12f);
      out[(size_t)t * TILE_ROWS + wave] = ph / denom;
    }

    __syncthreads();   // all reads of `buf` done before it is DMA-overwritten
    buf ^= 1;
  }
}

extern "C" void kernel_launch(void* const* d_in, const int* in_sizes, int n_in,
                              void* d_out, int out_size, void* d_ws, size_t ws_size,
                              hipStream_t stream) {
  (void)n_in; (void)out_size; (void)d_ws; (void)ws_size;
  const float* p   = (const float*)d_in[0];
  const float* h   = (const float*)d_in[1];
  float*       out = (float*)d_out;

  const int N     = in_sizes[0] / DCOLS;   // 65536 for the reference shapes
  const int tiles = N / TILE_ROWS;         // N divisible by 4 here
  int grid = tiles < 4096 ? tiles : 4096;  // 4 pipelined tiles per block
  if (grid < 1) grid = 1;

  CosineSimilarity_76785425318088_kernel<<<dim3(grid), dim3(BLOCK), 0, stream>>>(
      p, h, out, tiles);
}